// RNNModel_43679817400358
// MI455X (gfx1250) — compile-verified
//
#include <hip/hip_runtime.h>

// MI455X / gfx1250 — wave32, WMMA 16x16x4 f32, async global->LDS prefetch.
// 4-layer ReLU RNN, fused input-GEMM + recurrence scan.

typedef __attribute__((ext_vector_type(2))) float v2f;
typedef __attribute__((ext_vector_type(4))) float v4f;
typedef __attribute__((ext_vector_type(8))) float v8f;
typedef __attribute__((ext_vector_type(4))) int   v4i;

typedef __attribute__((address_space(1))) v4i gv4i;  // global-AS 4xi32
typedef __attribute__((address_space(3))) v4i lv4i;  // LDS-AS    4xi32

#define B_ 128
#define T_ 2048
#define H_ 128

#if defined(__has_builtin)
#if __has_builtin(__builtin_amdgcn_global_load_async_to_lds_b128)
#define ASYNC_LDS 1
#endif
#endif

// Copy 32 contiguous bytes global->LDS for this lane (async when available).
// INST_OFFSET applies to both global and LDS addresses (ISA 15.18.3).
__device__ __forceinline__ void stage32B(const float* g, float* l) {
#ifdef ASYNC_LDS
    __builtin_amdgcn_global_load_async_to_lds_b128((gv4i*)g, (lv4i*)l, 0, 0);
    __builtin_amdgcn_global_load_async_to_lds_b128((gv4i*)g, (lv4i*)l, 16, 0);
#else
    *(v4f*)l       = *(const v4f*)g;
    *(v4f*)(l + 4) = *(const v4f*)(g + 4);
#endif
}

// Wait until at most the newest prefetch (2 async loads) is outstanding.
__device__ __forceinline__ void wait_prev_stage() {
#ifdef ASYNC_LDS
#if __has_builtin(__builtin_amdgcn_s_wait_asynccnt)
    __builtin_amdgcn_s_wait_asynccnt(2);
#else
    asm volatile("s_wait_asynccnt 0x2" ::: "memory");
#endif
#endif
}

// One block = 16 batch rows, 8 waves (256 threads); wave w owns output cols [16w,16w+16).
//
// fp32 WMMA 16x16x4 VGPR layouts (ISA 7.12.2):
//   A (16x4):  lane<16: v0=K0,v1=K1 row M=lane;  lane>=16: v0=K2,v1=K3 row M=lane-16
//   B (4x16):  lane<16: v0=K0,v1=K1 col N=lane;  lane>=16: v0=K2,v1=K3 col N=lane-16
//   C/D:       vgpr r: lanes0-15 -> (M=r, N=lane); lanes16-31 -> (M=r+8, N=lane-16)
template <int F>
__global__ __launch_bounds__(256) void rnn_layer_scan(
    const float* __restrict__ in_seq,  // [B, T, F]
    const float* __restrict__ w_ih,    // [H, F]
    const float* __restrict__ w_hh,    // [H, H]
    const float* __restrict__ b_ih,    // [H]
    const float* __restrict__ b_hh,    // [H]
    float* __restrict__ out_seq)       // [B, T, H]
{
    __shared__ float lds_h[16][H_];
    __shared__ float lds_in[2][16][(F == 128) ? 128 : 4];  // double-buffered input tile

    const int tid  = threadIdx.x;
    const int wid  = tid >> 5;
    const int lane = tid & 31;
    const int nl   = lane & 15;
    const int hi   = lane >> 4;
    const int koff = hi * 2;
    const int n    = wid * 16 + nl;    // global output column
    const int b0   = blockIdx.x * 16;  // first batch row of this block
    const int m    = nl;               // A-matrix row for this lane

    // ---- per-wave weight fragments resident in VGPRs for the whole scan
    v2f wh[32];
#pragma unroll
    for (int kc = 0; kc < 32; ++kc) {
        const float* p = w_hh + (size_t)n * H_ + kc * 4 + koff;
        wh[kc] = v2f{p[0], p[1]};
    }
    v2f   wi[(F == 128) ? 32 : 1];
    float wi0 = 0.f, wi1 = 0.f;
    if constexpr (F == 128) {
#pragma unroll
        for (int kc = 0; kc < 32; ++kc) {
            const float* p = w_ih + (size_t)n * F + kc * 4 + koff;
            wi[kc] = v2f{p[0], p[1]};
        }
    } else {
        wi0 = w_ih[n * F + 0];
        wi1 = w_ih[n * F + 1];
    }
    const float bias = b_ih[n] + b_hh[n];

    // ---- staging coordinates / t=0 prefetch
    const int    mr  = tid >> 4;        // staged row 0..15
    const int    k0s = (tid & 15) * 8;  // staged col base
    const float* src_base = nullptr;
    if constexpr (F == 128) {
        src_base = in_seq + (size_t)(b0 + mr) * T_ * F + k0s;
        stage32B(src_base, &lds_in[0][mr][k0s]);
    }
    v2f xc[8];  // F==2: register-pipelined inputs
    if constexpr (F == 2) {
#pragma unroll
        for (int r = 0; r < 8; ++r)
            xc[r] = *(const v2f*)(in_seq + (size_t)(b0 + r + 8 * hi) * T_ * 2);
    }

    // ---- h0 = 0
    for (int i = tid; i < 16 * H_; i += 256) ((float*)lds_h)[i] = 0.f;
    __syncthreads();

    for (int t = 0; t < T_; ++t) {
        const int tn = (t + 1 < T_) ? (t + 1) : t;
        v2f xn[8];
        if constexpr (F == 128) {
            // prefetch tile t+1 into the other LDS buffer (async, stays in flight)
            stage32B(src_base + (size_t)tn * F, &lds_in[(t + 1) & 1][mr][k0s]);
            wait_prev_stage();  // tile t has landed (in-order retirement)
        } else {
#pragma unroll
            for (int r = 0; r < 8; ++r)
                xn[r] = *(const v2f*)(in_seq + ((size_t)(b0 + r + 8 * hi) * T_ + tn) * 2);
        }
        __syncthreads();  // tile t visible to all waves; prev h writes visible

        v8f acc;
#pragma unroll
        for (int r = 0; r < 8; ++r) acc[r] = bias;

        // input contribution (independent of the serial chain)
        if constexpr (F == 128) {
            const float* at = &lds_in[t & 1][m][0];
#pragma unroll
            for (int kc = 0; kc < 32; ++kc) {
                v2f a = *(const v2f*)(at + kc * 4 + koff);
                acc = __builtin_amdgcn_wmma_f32_16x16x4_f32(
                    false, a, false, wi[kc], (short)0, acc, false, false);
            }
        } else {
#pragma unroll
            for (int r = 0; r < 8; ++r) acc[r] += xc[r].x * wi0 + xc[r].y * wi1;
        }

        // recurrent contribution: h_{t-1} @ W_hh^T
#pragma unroll
        for (int kc = 0; kc < 32; ++kc) {
            v2f a = *(const v2f*)&lds_h[m][kc * 4 + koff];
            acc = __builtin_amdgcn_wmma_f32_16x16x4_f32(
                false, a, false, wh[kc], (short)0, acc, false, false);
        }

        __syncthreads();  // all waves done reading lds_h / lds_in[t&1]

        // relu; h_t -> LDS (next step) and -> global sequence
#pragma unroll
        for (int r = 0; r < 8; ++r) {
            float v    = acc[r] < 0.f ? 0.f : acc[r];
            int   brow = r + 8 * hi;
            lds_h[brow][n] = v;
            out_seq[((size_t)(b0 + brow) * T_ + t) * H_ + n] = v;
        }
        if constexpr (F == 2) {
#pragma unroll
            for (int r = 0; r < 8; ++r) xc[r] = xn[r];
        }
    }
}

// Final FC on last timestep: out[b] = h[b,T-1,:] . fc_w[0,:] + fc_b[0]   (C == 1)
__global__ __launch_bounds__(128) void fc_last(
    const float* __restrict__ h_seq,
    const float* __restrict__ fc_w,
    const float* __restrict__ fc_b,
    float* __restrict__ out)
{
    int b = blockIdx.x * blockDim.x + threadIdx.x;
    if (b < B_) {
        const float* h = h_seq + ((size_t)b * T_ + (T_ - 1)) * H_;
        float s = fc_b[0];
#pragma unroll 4
        for (int k = 0; k < H_; ++k) s += h[k] * fc_w[k];
        out[b] = s;
    }
}

extern "C" void kernel_launch(void* const* d_in, const int* in_sizes, int n_in,
                              void* d_out, int out_size, void* d_ws, size_t ws_size,
                              hipStream_t stream) {
    (void)in_sizes; (void)n_in; (void)out_size; (void)ws_size;
    const float* x         = (const float*)d_in[0];  // [B,T,I]
    const float* w_ih0     = (const float*)d_in[1];  // [H,I]
    const float* w_ih_rest = (const float*)d_in[2];  // [L-1,H,H]
    const float* w_hh      = (const float*)d_in[3];  // [L,H,H]
    const float* b_ih      = (const float*)d_in[4];  // [L,H]
    const float* b_hh      = (const float*)d_in[5];  // [L,H]
    const float* fc_w      = (const float*)d_in[6];  // [C,H]
    const float* fc_b      = (const float*)d_in[7];  // [C]
    float*       out       = (float*)d_out;          // [B]

    const size_t seq_elems = (size_t)B_ * T_ * H_;
    float* buf0 = (float*)d_ws;
    float* buf1 = buf0 + seq_elems;

    dim3 grid(B_ / 16), block(256);

    rnn_layer_scan<2><<<grid, block, 0, stream>>>(
        x, w_ih0, w_hh + 0 * H_ * H_, b_ih + 0 * H_, b_hh + 0 * H_, buf0);
    rnn_layer_scan<128><<<grid, block, 0, stream>>>(
        buf0, w_ih_rest + 0 * H_ * H_, w_hh + 1 * H_ * H_, b_ih + 1 * H_, b_hh + 1 * H_, buf1);
    rnn_layer_scan<128><<<grid, block, 0, stream>>>(
        buf1, w_ih_rest + 1 * H_ * H_, w_hh + 2 * H_ * H_, b_ih + 2 * H_, b_hh + 2 * H_, buf0);
    rnn_layer_scan<128><<<grid, block, 0, stream>>>(
        buf0, w_ih_rest + 2 * H_ * H_, w_hh + 3 * H_ * H_, b_ih + 3 * H_, b_hh + 3 * H_, buf1);

    fc_last<<<1, 128, 0, stream>>>(buf1, fc_w, fc_b, out);
}